// TP_nonlinearity_81870666596813
// MI455X (gfx1250) — compile-verified
//
#include <hip/hip_runtime.h>

// CDNA5 / gfx1250 wave32 WMMA fp32 path, fully specialized per (l1,l2,l3) combo.
typedef __attribute__((ext_vector_type(2))) float v2f;
typedef __attribute__((ext_vector_type(8))) float v8f;

#define NCOMBO 14

struct Params {
    const float* w[NCOMBO];   // w3j pointers in setup_inputs order
};

// Per-l3 output channel counts and prefix (in units of B) of output offsets:
//   obase[l3] = B * OPRE[l3]   (floats)
// OPRE prefix over l3 of OCH[l3]*(2*l3+1): {0,12288,61440,143360,200704}

template<int L1, int L2, int L3, int PBLK>
__device__ __forceinline__
void tp_combo(float* sA, float* sT,
              const float* __restrict__ xA_base,
              const float* __restrict__ xB_base,
              const float* __restrict__ W,
              float* __restrict__ out, int B)
{
    constexpr int dimA = 2 * L1 + 1;          // GEMM K (components k of x_l1)
    constexpr int dimB = 2 * L2 + 1;
    constexpr int dimM = 2 * L3 + 1;
    constexpr int Kpad = (L1 == 2) ? 8 : 4;
    constexpr int Ncols = 64 * dimM;          // GEMM N == output row stride
    constexpr long long OPRE[5] = {0, 12288, 61440, 143360, 200704};
    constexpr int OCH[5] = {12288, 16384, 16384, 8192, 4096};

    const int b   = blockIdx.x;
    const int tid = threadIdx.x;
    const float* __restrict__ xA = xA_base + (size_t)b * 64 * dimA;
    const float* __restrict__ xB = xB_base + (size_t)b * 64 * dimB;

    // ---- stage A (64 x 8, K zero-padded, stride 8) ----
    #pragma unroll
    for (int base = 0; base < 64 * 8; base += 256) {
        const int idx = base + tid;
        const int i = idx >> 3, k = idx & 7;
        sA[idx] = (k < dimA) ? xA[i * dimA + k] : 0.0f;
    }
    // ---- stage t[k, j*dimM+M] = sum_n w3j[k,n,M] * x_l2[b,j,n] ----
    constexpr int TCNT = Kpad * Ncols;
    for (int idx = tid; idx < TCNT; idx += 256) {
        const int k   = idx / Ncols;          // constant divisors -> cheap
        const int col = idx - k * Ncols;
        const int j   = col / dimM;
        const int M   = col - j * dimM;
        float acc = 0.0f;
        if (k < dimA) {
            #pragma unroll
            for (int n = 0; n < dimB; ++n)
                acc += W[(k * dimB + n) * dimM + M] * xB[j * dimB + n];
        }
        sT[idx] = acc;
    }
    __syncthreads();

    // output block for (b, pair, l3): contiguous row-major 64 x Ncols matrix
    float* __restrict__ outBlk = out + (size_t)B * OPRE[L3]
        + (size_t)b * OCH[L3] * dimM + (size_t)PBLK * 4096 * dimM;

    const int wave    = tid >> 5;
    const int lane    = tid & 31;
    const int halfsel = lane >> 4;
    const int l16     = lane & 15;

    constexpr int numTilesN = Ncols / 16;     // 4*dimM
    constexpr int numTiles  = 4 * numTilesN;

    // wave-uniform tile loop: EXEC all-ones around every WMMA
    for (int t = wave; t < numTiles; t += 8) {
        const int tm   = t / numTilesN;       // constant divisor
        const int tn   = t - tm * numTilesN;
        const int arow = tm * 16 + l16;
        const int bcol = tn * 16 + l16;

        v8f acc = {0.f,0.f,0.f,0.f,0.f,0.f,0.f,0.f};
        #pragma unroll
        for (int kc = 0; kc < Kpad; kc += 4) {
            // 32-bit A 16x4 layout: lanes0-15 -> (K0,K1), lanes16-31 -> (K2,K3)
            const int kl = kc + 2 * halfsel;
            v2f a, bb;
            a.x  = sA[arow * 8 + kl];
            a.y  = sA[arow * 8 + kl + 1];
            // B 4x16: rows striped across lanes within one VGPR (mirrors A halves)
            bb.x = sT[kl * Ncols + bcol];
            bb.y = sT[(kl + 1) * Ncols + bcol];
            acc = __builtin_amdgcn_wmma_f32_16x16x4_f32(
                false, a, false, bb, (short)0, acc, false, false);
        }

        // C/D 16x16 layout: VGPR v <-> row v (lanes0-15) / v+8 (lanes16-31)
        // One base address + compile-time immediate row offsets; NT stores:
        // 486 MB streamed output never re-read, keep it out of L2.
        float* __restrict__ pr =
            outBlk + (size_t)(tm * 16 + 8 * halfsel) * Ncols + bcol;
        #pragma unroll
        for (int v = 0; v < 8; ++v)
            __builtin_nontemporal_store(acc[v], pr + v * Ncols);
    }
}

__global__ __launch_bounds__(256)
void tp_wmma_kernel(const float* __restrict__ x0,
                    const float* __restrict__ x1,
                    const float* __restrict__ x2,
                    Params p, float* __restrict__ out, int B)
{
    __shared__ float sA[64 * 8];       // 2 KB
    __shared__ float sT[8 * 64 * 9];   // 18 KB

    switch (blockIdx.y) {              // wave-uniform dispatch, 14 specializations
    case 0:  tp_combo<0,0,0,0>(sA, sT, x0, x0, p.w[0],  out, B); break;
    case 1:  tp_combo<0,1,1,0>(sA, sT, x0, x1, p.w[1],  out, B); break;
    case 2:  tp_combo<0,2,2,0>(sA, sT, x0, x2, p.w[2],  out, B); break;
    case 3:  tp_combo<1,1,0,1>(sA, sT, x1, x1, p.w[3],  out, B); break;
    case 4:  tp_combo<1,1,1,1>(sA, sT, x1, x1, p.w[4],  out, B); break;
    case 5:  tp_combo<1,1,2,1>(sA, sT, x1, x1, p.w[5],  out, B); break;
    case 6:  tp_combo<1,2,1,2>(sA, sT, x1, x2, p.w[6],  out, B); break;
    case 7:  tp_combo<1,2,2,2>(sA, sT, x1, x2, p.w[7],  out, B); break;
    case 8:  tp_combo<1,2,3,0>(sA, sT, x1, x2, p.w[8],  out, B); break;
    case 9:  tp_combo<2,2,0,2>(sA, sT, x2, x2, p.w[9],  out, B); break;
    case 10: tp_combo<2,2,1,3>(sA, sT, x2, x2, p.w[10], out, B); break;
    case 11: tp_combo<2,2,2,3>(sA, sT, x2, x2, p.w[11], out, B); break;
    case 12: tp_combo<2,2,3,1>(sA, sT, x2, x2, p.w[12], out, B); break;
    case 13: tp_combo<2,2,4,0>(sA, sT, x2, x2, p.w[13], out, B); break;
    default: break;
    }
}

extern "C" void kernel_launch(void* const* d_in, const int* in_sizes, int n_in,
                              void* d_out, int out_size, void* d_ws, size_t ws_size,
                              hipStream_t stream) {
    const float* x0 = (const float*)d_in[0];
    const float* x1 = (const float*)d_in[1];
    const float* x2 = (const float*)d_in[2];

    Params p;
    for (int i = 0; i < NCOMBO; ++i) p.w[i] = (const float*)d_in[3 + i];

    const int B = in_sizes[0] / 64;   // x0 is [B, 64, 1]

    dim3 grid(B, NCOMBO);
    tp_wmma_kernel<<<grid, 256, 0, stream>>>(x0, x1, x2, p, (float*)d_out, B);
}